// HalfedgeToVertexLayer_20736102105650
// MI455X (gfx1250) — compile-verified
//
#include <hip/hip_runtime.h>

// Native clang vector types.
typedef float v4f __attribute__((ext_vector_type(4)));
typedef int v4i __attribute__((vector_size(16)));  // matches builtin param type

#define WAVE 32
#define BLOCK 256
#define WPB (BLOCK / WAVE)  // 8 waves per block

// gfx1250 async global->LDS copy path (ASYNCcnt). Probe-confirmed on this
// toolchain; signature: (v4i AS1* src, v4i AS3* dst, imm offset, imm cpol)
#if defined(__has_builtin)
#if __has_builtin(__builtin_amdgcn_global_load_async_to_lds_b128) && \
    __has_builtin(__builtin_amdgcn_s_wait_asynccnt)
#define HTV_ASYNC 1
#else
#define HTV_ASYNC 0
#endif
#else
#define HTV_ASYNC 0
#endif

#define GAS1_V4I(p) ((__attribute__((address_space(1))) v4i*)(p))
#define LAS3_V4I(p) ((__attribute__((address_space(3))) v4i*)(p))

// CPol: TH in bits [2:0]; TH=1 -> non-temporal (the x stream is use-once, so
// keep it from evicting the L2-resident output working set).
#define CPOL_NT 1

// ---------------------------------------------------------------------------
// Scatter-add (placed first so the disasm snippet shows it).
// Wave layout: each wave consumes 4 half-edges per step.
//   lane l -> half-edge  base + (l >> 3)
//          -> features   [(l&7)*4 .. (l&7)*4+3]   (one float4)
// 32 lanes x 16B = one contiguous 512B block per wave.
// Async path: double-buffer the 512B tile through LDS with
// global_load_async_to_lds_b128 (NT), overlapping the HBM stream with atomic
// issue.  Atomics: native global_atomic_add_f32 scope:SCOPE_DEV, L2-resident
// (128MB out + 4MB valence < 192MB L2).
// ---------------------------------------------------------------------------
__global__ void __launch_bounds__(BLOCK) htv_scatter_kernel(
    const float* __restrict__ x, const int* __restrict__ vertex_ids,
    float* __restrict__ out, float* __restrict__ valence, int n_he) {
  const int lane = threadIdx.x & (WAVE - 1);
  const int gwave = (blockIdx.x * BLOCK + threadIdx.x) >> 5;
  const int n_waves = (gridDim.x * BLOCK) >> 5;
  const int sub = lane >> 3;       // which of the 4 half-edges
  const int fo = (lane & 7) << 2;  // feature offset: 0,4,...,28
  const int step = n_waves * 4;
  const int base0 = gwave * 4;

#if HTV_ASYNC
  __shared__ v4f stage[WPB][2][WAVE];  // 8KB: per-wave private double buffer
  const int w = threadIdx.x >> 5;

  if (base0 < n_he) {  // prologue: issue tile 0
    const int he0 = min(base0 + sub, n_he - 1);
    __builtin_amdgcn_global_load_async_to_lds_b128(
        GAS1_V4I(x + (size_t)he0 * 32 + fo), LAS3_V4I(&stage[w][0][lane]), 0,
        CPOL_NT);
  }
  int buf = 0;
  for (int base = base0; base < n_he; base += step) {
    const int nxt = base + step;  // uniform within the wave
    if (nxt < n_he) {
      const int hen = min(nxt + sub, n_he - 1);
      __builtin_amdgcn_global_load_async_to_lds_b128(
          GAS1_V4I(x + (size_t)hen * 32 + fo),
          LAS3_V4I(&stage[w][buf ^ 1][lane]), 0, CPOL_NT);
      __builtin_amdgcn_s_wait_asynccnt(1);  // tile `base` has landed
    } else {
      __builtin_amdgcn_s_wait_asynccnt(0);
    }
    const v4f v = stage[w][buf][lane];  // ds_load_b128
    const int he = base + sub;
    if (he < n_he) {
      const int vtx = __builtin_nontemporal_load(vertex_ids + he);
      float* dst = out + (size_t)vtx * 32 + fo;
      atomicAdd(dst + 0, v.x);
      atomicAdd(dst + 1, v.y);
      atomicAdd(dst + 2, v.z);
      atomicAdd(dst + 3, v.w);
      if ((lane & 7) == 0) atomicAdd(valence + vtx, 1.0f);
    }
    buf ^= 1;
  }
#else
  for (int base = base0; base < n_he; base += step) {
    const int he = base + sub;
    if (he < n_he) {
      const v4f v =
          __builtin_nontemporal_load((const v4f*)(x + (size_t)he * 32 + fo));
      const int vtx = __builtin_nontemporal_load(vertex_ids + he);
      float* dst = out + (size_t)vtx * 32 + fo;
      atomicAdd(dst + 0, v.x);
      atomicAdd(dst + 1, v.y);
      atomicAdd(dst + 2, v.z);
      atomicAdd(dst + 3, v.w);
      if ((lane & 7) == 0) atomicAdd(valence + vtx, 1.0f);
    }
  }
#endif
}

// ---------------------------------------------------------------------------
// Zero the accumulator (d_out, [V,32] f32) and valence ([V] f32).
// ---------------------------------------------------------------------------
__global__ void htv_zero_kernel(float* __restrict__ out, int n_out4,
                                float* __restrict__ valence, int n_val) {
  const int stride = gridDim.x * blockDim.x;
  const int tid = blockIdx.x * blockDim.x + threadIdx.x;
  v4f z = (v4f){0.0f, 0.0f, 0.0f, 0.0f};
  v4f* out4 = (v4f*)out;
  for (int i = tid; i < n_out4; i += stride) out4[i] = z;
  for (int i = tid; i < n_val; i += stride) valence[i] = 0.0f;
}

// ---------------------------------------------------------------------------
// Normalize: out[v][:] *= 1 / max(valence[v], 1).
// ---------------------------------------------------------------------------
__global__ void htv_normalize_kernel(float* __restrict__ out,
                                     const float* __restrict__ valence,
                                     int n_out4) {
  const int stride = gridDim.x * blockDim.x;
  v4f* out4 = (v4f*)out;
  for (int i = blockIdx.x * blockDim.x + threadIdx.x; i < n_out4; i += stride) {
    const int v = i >> 3;  // 8 float4s per 32-float vertex row
    const float r = 1.0f / fmaxf(valence[v], 1.0f);
    v4f d = out4[i];
    d *= r;
    out4[i] = d;
  }
}

// ---------------------------------------------------------------------------
// Launch
// ---------------------------------------------------------------------------
extern "C" void kernel_launch(void* const* d_in, const int* in_sizes, int n_in,
                              void* d_out, int out_size, void* d_ws,
                              size_t ws_size, hipStream_t stream) {
  const float* x = (const float*)d_in[0];
  const int* vertex_ids = (const int*)d_in[1];
  // d_in[2] is a device-side scalar (n_vertices); derive host-side instead:
  const int n_he = in_sizes[1];   // 6,000,000
  const int n_v = out_size / 32;  // 1,000,000
  const int n_out4 = out_size / 4;

  float* out = (float*)d_out;
  float* valence = (float*)d_ws;  // n_v * 4 bytes of scratch

  // 1) zero accumulators
  htv_zero_kernel<<<2048, BLOCK, 0, stream>>>(out, n_out4, valence, n_v);
  // 2) scatter-add (grid-stride; each wave eats 4 half-edges per iteration)
  htv_scatter_kernel<<<8192, BLOCK, 0, stream>>>(x, vertex_ids, out, valence,
                                                 n_he);
  // 3) normalize by valence
  htv_normalize_kernel<<<4096, BLOCK, 0, stream>>>(out, valence, n_out4);
}